// SoftSort_72430328480080
// MI455X (gfx1250) — compile-verified
//
#include <hip/hip_runtime.h>

// SoftSort, MI455X (gfx1250, wave32).
// Softmax shift-invariance => P rows identical => out[b,i,:] = (sum_j e_j F[j,:]) / Z.
// Total HBM traffic ~8.1 MB -> ~0.35us at 23.3 TB/s; purely bandwidth-bound.
// The weighted reduction is done on the CDNA5 matrix pipe with V_WMMA_F32_16X16X4_F32
// (exact f32, matches reference numerics; matrix pipe is free at this arithmetic intensity).

#define TAU_F 1.0f
constexpr int B_ = 4;
constexpr int N_ = 4096;
constexpr int C_ = 64;
constexpr int CHUNKS = 16;          // j-chunks per batch in the reduction pass
constexpr int JPB    = N_ / CHUNKS; // 256 rows of F per block

typedef __attribute__((ext_vector_type(2))) float v2f;
typedef __attribute__((ext_vector_type(4))) float v4f;
typedef __attribute__((ext_vector_type(8))) float v8f;

// d_ws float layout: [0, B*C)           unnormalized g accumulators (per batch row)
//                    [B*C, B*C+B)       per-batch max m
//                    [B*C+B, B*C+2B)    per-batch Z = sum exp((s-m)/tau)
#define WS_M_OFF (B_ * C_)
#define WS_Z_OFF (B_ * C_ + B_)

// ---------- Pass 1: per-batch max + Z; zero the g accumulators ----------
__global__ void softsort_stats(const float* __restrict__ scores,
                               float* __restrict__ ws) {
  __shared__ float red[256];
  const int b = blockIdx.x;
  const int t = threadIdx.x;
  const float* s = scores + b * N_;

  if (t < C_) ws[b * C_ + t] = 0.0f;   // zero accumulator row (ws is poisoned)

  float m = -3.402823466e38f;
  for (int j = t; j < N_; j += 256) m = fmaxf(m, s[j]);
  red[t] = m;
  __syncthreads();
  for (int off = 128; off > 0; off >>= 1) {
    if (t < off) red[t] = fmaxf(red[t], red[t + off]);
    __syncthreads();
  }
  m = red[0];
  __syncthreads();

  const float it = 1.0f / TAU_F;
  float z = 0.0f;
  for (int j = t; j < N_; j += 256) z += __expf((s[j] - m) * it);
  red[t] = z;
  __syncthreads();
  for (int off = 128; off > 0; off >>= 1) {
    if (t < off) red[t] += red[t + off];
    __syncthreads();
  }
  if (t == 0) {
    ws[WS_M_OFF + b] = m;
    ws[WS_Z_OFF + b] = red[0];
  }
}

// ---------- Pass 2: g[b,c] += sum_{j in chunk} e_j * F[j,c] via WMMA f32 ----------
// Block = 256 threads = 8 waves. Wave w: channel tile ct = w&3 (16 channels),
// j-strip = w>>2 (halves the chunk). A-matrix: e[j0..j0+3] replicated across all
// 16 M-rows (lanes 0-15: K=0,1 ; lanes 16-31: K=2,3). B-matrix: F[j0:j0+4, c0:c0+16].
// Row 0 of D (acc[0], lanes 0-15) then holds exact f32 partials of g[c0+n].
__global__ void softsort_wsum(const float* __restrict__ scores,
                              const float* __restrict__ feats,
                              float* __restrict__ ws) {
  __shared__ float e[JPB];
  __shared__ float part[8 * 16];

  const int b     = blockIdx.x / CHUNKS;
  const int chunk = blockIdx.x % CHUNKS;
  const int j0b   = chunk * JPB;
  const int t     = threadIdx.x;

  const float m  = ws[WS_M_OFF + b];
  const float it = 1.0f / TAU_F;

  // Stage softmax numerators for this chunk into LDS (one per thread).
  e[t] = __expf((scores[b * N_ + j0b + t] - m) * it);
  __syncthreads();

  const int w    = t >> 5;          // wave id 0..7
  const int lane = t & 31;
  const int hi   = (lane >> 4) & 1; // 0: lanes 0-15, 1: lanes 16-31
  const int n    = lane & 15;       // B/D column within tile
  const int ct   = w & 3;           // channel tile (16 channels)
  const int strip = w >> 2;         // j half of the chunk
  const int c0   = ct * 16;

  const float* F = feats + (size_t)b * N_ * C_;

  v8f acc = {};
  const int jlo = strip * (JPB / 2);
  const int jhi = jlo + (JPB / 2);
  for (int j = jlo; j < jhi; j += 4) {
    const int jk = j0b + j + 2 * hi;          // K row handled by this lane half
    v2f a;                                    // A[m,k] = e (replicated over m)
    a.x = e[j + 2 * hi];                      // K = 0 (lo) / 2 (hi)
    a.y = e[j + 2 * hi + 1];                  // K = 1 (lo) / 3 (hi)
    const float* Frow = F + (size_t)jk * C_ + c0 + n;
    v2f bm;                                   // B[k,n] = F[jk, c0+n]
    bm.x = Frow[0];                           // K = 0 / 2
    bm.y = Frow[C_];                          // K = 1 / 3
    // D = A x B + C  (8 args: neg_a, A, neg_b, B, c_mod, C, reuse_a, reuse_b)
    acc = __builtin_amdgcn_wmma_f32_16x16x4_f32(false, a, false, bm,
                                                (short)0, acc, false, false);
  }

  // D row 0 lives in acc[0] of lanes 0..15: partial g[c0+n] for this wave.
  if (hi == 0) part[w * 16 + n] = acc[0];
  __syncthreads();

  if (t < C_) {
    const int cct = t >> 4, ln = t & 15;
    const float g = part[cct * 16 + ln] + part[(cct + 4) * 16 + ln];
    atomicAdd(&ws[b * C_ + t], g);            // global_atomic_add_f32
  }
}

// ---------- Pass 3: broadcast out[b,i,:] = g[b,:] / Z[b], float4 stores ----------
__global__ void softsort_bcast(const float* __restrict__ ws,
                               float* __restrict__ out) {
  const size_t idx   = (size_t)blockIdx.x * blockDim.x + threadIdx.x; // float4 index
  const size_t total = (size_t)B_ * N_ * C_ / 4;
  if (idx >= total) return;
  const int per_b = N_ * C_ / 4;
  const int b  = (int)(idx / per_b);
  const int c4 = (int)(idx & (C_ / 4 - 1));
  const float invZ = 1.0f / ws[WS_Z_OFF + b];
  const v4f* row = (const v4f*)(ws + b * C_);
  v4f v = row[c4];
  v.x *= invZ; v.y *= invZ; v.z *= invZ; v.w *= invZ;
  ((v4f*)out)[idx] = v;                        // global_store_b128
}

extern "C" void kernel_launch(void* const* d_in, const int* in_sizes, int n_in,
                              void* d_out, int out_size, void* d_ws, size_t ws_size,
                              hipStream_t stream) {
  const float* scores = (const float*)d_in[0]; // (B,N,1) flat
  const float* feats  = (const float*)d_in[1]; // (B,N,C) flat
  float* out = (float*)d_out;
  float* wsf = (float*)d_ws;

  softsort_stats<<<B_, 256, 0, stream>>>(scores, wsf);
  softsort_wsum<<<B_ * CHUNKS, 256, 0, stream>>>(scores, feats, wsf);
  const int tot4 = B_ * N_ * C_ / 4;
  softsort_bcast<<<(tot4 + 255) / 256, 256, 0, stream>>>(wsf, out);
}